// QuantumBilinearController_68143951118358
// MI455X (gfx1250) — compile-verified
//
#include <hip/hip_runtime.h>

#define NTC 1000
#define DT0C 0.02f

typedef float v2f __attribute__((ext_vector_type(2)));
typedef float v8f __attribute__((ext_vector_type(8)));

// assume-non-null helpers: kill generic->LDS addrspacecast null guards
static __device__ inline void nn1(const void* a){ __builtin_assume(a != nullptr); }
static __device__ inline void nn2(const void* a, const void* b){ nn1(a); nn1(b); }
static __device__ inline void nn3(const void* a, const void* b, const void* c){ nn2(a,b); nn1(c); }

// ---------------- workspace layout (float elements) ----------------
#define WS_RI     0                          // 64   : R^{-1}
#define WS_BRI    64                         // 512  : B @ Ri
#define WS_BRIBT  1024                       // 4096 : B Ri B^T
#define WS_PS     8192                       // (NTC+1)*4096 : LQR Riccati trajectory Ps
#define WS_PK1    (WS_PS  + (NTC+1)*4096)    // (NTC+1)*4096 : P_k1 trajectory
#define WS_XG     (WS_PK1 + (NTC+1)*4096)    // NTC*64 : x_grid
#define WS_XK1    (WS_XG  + NTC*64)          // NTC*64 : x_k1
#define WS_UG     (WS_XK1 + NTC*64)          // NTC*8  : u_grid
#define WS_TMP    (WS_UG  + NTC*8)           // scratch matrices
#define TM(n)     (WS_TMP + (n)*4096)

#define OUT_X 0
#define OUT_U (3*NTC*64)

// ---------------- async global -> LDS copy (CDNA5) -----------------
// 16KB (4096 floats) per call; 256 threads x 4 iterations x b128.
// LDS address for the async op = low 32 bits of the generic pointer.
static __device__ inline void asyncCopy16k(float* dstLds, const float* src)
{
  nn2(dstLds, src);
  const unsigned base = (unsigned)(uintptr_t)dstLds;
  #pragma unroll
  for (int it = 0; it < 4; ++it) {
    const int idx = threadIdx.x*4 + it*1024;
    const unsigned lo = base + (unsigned)idx*4u;
    const unsigned long long ga = (unsigned long long)(uintptr_t)(src + idx);
    asm volatile("global_load_async_to_lds_b128 %0, %1, off" :: "v"(lo), "v"(ga) : "memory");
  }
}
static __device__ inline void asyncWait()
{
  asm volatile("s_wait_asynccnt 0" ::: "memory");
}

// ---------------- 64x64 matmul on the matrix pipes -----------------
// D = alpha * op(A) @ op(B) + beta * D   (row-major 64x64)
// 16 tiles of 16x16, K=64 via 16 chained V_WMMA_F32_16X16X4_F32.
static __device__ inline void mm64(float* __restrict__ D,
                                   const float* __restrict__ A,
                                   const float* __restrict__ B,
                                   bool tA, bool tB, float alpha, float beta)
{
  nn3(D, A, B);
  const int tid  = threadIdx.x;
  const int wave = tid >> 5;
  const int lane = tid & 31;
  const int l15  = lane & 15;
  const int hi2  = (lane >= 16) ? 2 : 0;
  const int hi8  = (lane >= 16) ? 8 : 0;
  __syncthreads();
  #pragma unroll
  for (int t = 0; t < 2; ++t) {
    const int tile = wave + t*8;
    const int bi = tile >> 2, bj = tile & 3;
    const int rowA  = bi*16 + l15;
    const int colB  = bj*16 + l15;
    const int rowD0 = bi*16 + hi8;
    const int colD  = bj*16 + l15;
    v8f acc;
    if (beta != 0.0f) {
      #pragma unroll
      for (int v = 0; v < 8; ++v) acc[v] = beta * D[(rowD0+v)*64 + colD];
    } else {
      #pragma unroll
      for (int v = 0; v < 8; ++v) acc[v] = 0.0f;
    }
    #pragma unroll
    for (int kk = 0; kk < 64; kk += 4) {
      const int k0 = kk + hi2;
      v2f a, b;
      a.x = alpha * (tA ? A[(k0+0)*64 + rowA] : A[rowA*64 + (k0+0)]);
      a.y = alpha * (tA ? A[(k0+1)*64 + rowA] : A[rowA*64 + (k0+1)]);
      b.x = tB ? B[colB*64 + (k0+0)] : B[(k0+0)*64 + colB];
      b.y = tB ? B[colB*64 + (k0+1)] : B[(k0+1)*64 + colB];
      acc = __builtin_amdgcn_wmma_f32_16x16x4_f32(false, a, false, b, (short)0, acc, false, false);
    }
    #pragma unroll
    for (int v = 0; v < 8; ++v) D[(rowD0+v)*64 + colD] = acc[v];
  }
  __syncthreads();
}

// ---------------- elementwise / small helpers (blockDim == 256) ----
static __device__ inline void ew_stage(float* yt, const float* y0, const float* k, float c){
  nn3(yt, y0, k);
  __syncthreads();
  for (int i = threadIdx.x; i < 4096; i += 256) yt[i] = y0[i] + c*k[i];
  __syncthreads();
}
static __device__ inline void ew_accum(float* acc, const float* k, float w, bool init){
  nn2(acc, k);
  __syncthreads();
  for (int i = threadIdx.x; i < 4096; i += 256) acc[i] = (init ? 0.0f : acc[i]) + w*k[i];
  __syncthreads();
}
static __device__ inline void mv64(float* y, const float* M, const float* x, bool tM, float alpha, float beta){
  nn3(y, M, x);
  __syncthreads();
  const int tid = threadIdx.x;
  if (tid < 64) {
    float s = 0.0f;
    for (int j = 0; j < 64; ++j) s += (tM ? M[j*64+tid] : M[tid*64+j]) * x[j];
    y[tid] = alpha*s + (beta == 0.0f ? 0.0f : beta*y[tid]);
  }
  __syncthreads();
}
static __device__ inline void v_stage(float* yt, const float* y0, const float* k, float c){
  nn3(yt, y0, k);
  __syncthreads();
  if (threadIdx.x < 64) yt[threadIdx.x] = y0[threadIdx.x] + c*k[threadIdx.x];
  __syncthreads();
}
static __device__ inline void v_accum(float* acc, const float* k, float w, bool init){
  nn2(acc, k);
  __syncthreads();
  if (threadIdx.x < 64) acc[threadIdx.x] = (init ? 0.0f : acc[threadIdx.x]) + w*k[threadIdx.x];
  __syncthreads();
}
// Hx[j,l] = sum_n H[j,n,l] x[n]
static __device__ inline void h_mid(float* Hx, const float* Hg, const float* x){
  nn3(Hx, Hg, x);
  __syncthreads();
  for (int idx = threadIdx.x; idx < 512; idx += 256) {
    const int j = idx >> 3, l = idx & 7;
    float s = 0.0f;
    for (int n = 0; n < 64; ++n) s += Hg[j*512 + n*8 + l] * x[n];
    Hx[idx] = s;
  }
  __syncthreads();
}
// Ep[i,k] = sum_m p[m] H[m,i,k]
static __device__ inline void h_first(float* Ep, const float* Hg, const float* p){
  nn3(Ep, Hg, p);
  __syncthreads();
  for (int idx = threadIdx.x; idx < 512; idx += 256) {
    float s = 0.0f;
    for (int m = 0; m < 64; ++m) s += p[m] * Hg[m*512 + idx];
    Ep[idx] = s;
  }
  __syncthreads();
}
// D = alpha * (sum_k H[i,j,k] d[k]) + beta*D
static __device__ inline void h_last(float* D, const float* Hg, const float* d, float alpha, float beta){
  nn3(D, Hg, d);
  __syncthreads();
  for (int idx = threadIdx.x; idx < 4096; idx += 256) {
    float s = 0.0f;
    #pragma unroll
    for (int k = 0; k < 8; ++k) s += Hg[idx*8 + k] * d[k];
    D[idx] = alpha*s + (beta == 0.0f ? 0.0f : beta*D[idx]);
  }
  __syncthreads();
}
// D(64x8) = X(64x8) @ Rm(8x8)
static __device__ inline void mk8_XR(float* D, const float* X, const float* Rm){
  nn3(D, X, Rm);
  __syncthreads();
  for (int idx = threadIdx.x; idx < 512; idx += 256) {
    const int i = idx >> 3, l = idx & 7;
    float s = 0.0f;
    #pragma unroll
    for (int r = 0; r < 8; ++r) s += X[i*8+r] * Rm[r*8+l];
    D[idx] = s;
  }
  __syncthreads();
}
// D(64x64) = alpha * X(64x8) @ Y(64x8)^T + beta*D
static __device__ inline void mk8_XYt(float* D, const float* X, const float* Y, float alpha, float beta){
  nn3(D, X, Y);
  __syncthreads();
  for (int idx = threadIdx.x; idx < 4096; idx += 256) {
    const int i = idx >> 6, j = idx & 63;
    float s = 0.0f;
    #pragma unroll
    for (int r = 0; r < 8; ++r) s += X[i*8+r] * Y[j*8+r];
    D[idx] = alpha*s + (beta == 0.0f ? 0.0f : beta*D[idx]);
  }
  __syncthreads();
}
// half-step-index interpolated P load into LDS via async DMA
// h = 2*grid_index (grid point) or odd (midpoint average of two tiles)
static __device__ inline void loadP_lds(float* dst, float* stage, const float* Pg, int h){
  nn3(dst, stage, Pg);
  __syncthreads();
  const float* a = Pg + (size_t)(h >> 1) * 4096;
  if (h & 1) {
    asyncCopy16k(dst, a);
    asyncCopy16k(stage, a + 4096);
    asyncWait();
    __syncthreads();
    for (int i = threadIdx.x; i < 4096; i += 256) dst[i] = 0.5f*(dst[i] + stage[i]);
  } else {
    asyncCopy16k(dst, a);
    asyncWait();
  }
  __syncthreads();
}
static __device__ inline void loadX(float* dst, const float* xg, int h){
  nn2(dst, xg);
  __syncthreads();
  const int t = threadIdx.x;
  if (t < 64) {
    if (h >= 2*(NTC-1))    dst[t] = xg[(NTC-1)*64 + t];
    else if ((h & 1) == 0) dst[t] = xg[(h >> 1)*64 + t];
    else                   dst[t] = 0.5f*(xg[(h >> 1)*64 + t] + xg[((h >> 1)+1)*64 + t]);
  }
  __syncthreads();
}
static __device__ inline void loadU(float* dst, const float* ug, int h){
  nn2(dst, ug);
  __syncthreads();
  const int t = threadIdx.x;
  if (t < 8) {
    if (h >= 2*(NTC-1))    dst[t] = ug[(NTC-1)*8 + t];
    else if ((h & 1) == 0) dst[t] = ug[(h >> 1)*8 + t];
    else                   dst[t] = 0.5f*(ug[(h >> 1)*8 + t] + ug[((h >> 1)+1)*8 + t]);
  }
  __syncthreads();
}

// A_k_tilde(p) = A - 0.5*(sum_k H[:,:,k] d[k]) - 0.5*(B Ri) Ep^T,
// d = Ri (B^T p), Ep[i,k] = sum_m p[m] H[m,i,k]
static __device__ inline void build_Akt_tilde(float* Akt, const float* Hg, const float* Ag,
                                              const float* sB, const float* sBRi, const float* sRi,
                                              const float* pk, float* Ep, float* cd)
{
  nn3(Akt, Hg, Ag); nn3(sB, sBRi, sRi); nn3(pk, Ep, cd);
  __syncthreads();
  if (threadIdx.x < 8) {
    const int l = threadIdx.x;
    float s = 0.0f;
    for (int m = 0; m < 64; ++m) s += sB[m*8 + l] * pk[m];
    cd[l] = s;
  }
  __syncthreads();
  if (threadIdx.x < 8) {
    const int k = threadIdx.x;
    float s = 0.0f;
    #pragma unroll
    for (int l = 0; l < 8; ++l) s += sRi[k*8 + l] * cd[l];
    cd[8 + k] = s;
  }
  __syncthreads();
  h_first(Ep, Hg, pk);
  h_last(Akt, Hg, cd + 8, -0.5f, 0.0f);
  __syncthreads();
  for (int i = threadIdx.x; i < 4096; i += 256) Akt[i] += Ag[i];
  __syncthreads();
  mk8_XYt(Akt, sBRi, Ep, -0.5f, 1.0f);
}
// S_k_tilde(x) = Bt Ri Bt^T - 0.5*(Hx Ri B^T + B Ri Hx^T), Bt = B + Hx
static __device__ inline void build_S(float* S, const float* Hg,
                                      const float* sB, const float* sBRi, const float* sRi,
                                      const float* xk, float* Hx, float* Bt, float* Us)
{
  nn3(S, Hg, sB); nn3(sBRi, sRi, xk); nn3(Hx, Bt, Us);
  h_mid(Hx, Hg, xk);
  __syncthreads();
  for (int i = threadIdx.x; i < 512; i += 256) Bt[i] = sB[i] + Hx[i];
  __syncthreads();
  mk8_XR(Us, Bt, sRi);
  mk8_XYt(S, Us, Bt, 1.0f, 0.0f);
  mk8_XR(Us, Hx, sRi);
  mk8_XYt(S, Us, sB, -0.5f, 1.0f);
  mk8_XYt(S, sBRi, Hx, -0.5f, 1.0f);
}

// ---------------- kernel 1: Ri, B Ri, B Ri B^T ---------------------
__global__ void __launch_bounds__(256) k_prep(const float* Rg, const float* Bg, float* ws)
{
  __shared__ float sRi[64];
  const int tid = threadIdx.x;
  if (tid == 0) {
    float M[8][16];
    for (int i = 0; i < 8; ++i)
      for (int j = 0; j < 8; ++j) { M[i][j] = Rg[i*8+j]; M[i][8+j] = (i == j) ? 1.0f : 0.0f; }
    for (int c = 0; c < 8; ++c) {
      const float inv = 1.0f / M[c][c];
      for (int j = 0; j < 16; ++j) M[c][j] *= inv;
      for (int r = 0; r < 8; ++r) if (r != c) {
        const float f = M[r][c];
        for (int j = 0; j < 16; ++j) M[r][j] -= f * M[c][j];
      }
    }
    for (int i = 0; i < 8; ++i) for (int j = 0; j < 8; ++j) sRi[i*8+j] = M[i][8+j];
  }
  __syncthreads();
  if (tid < 64) ws[WS_RI + tid] = sRi[tid];
  for (int idx = tid; idx < 512; idx += 256) {
    const int i = idx >> 3, l = idx & 7;
    float s = 0.0f;
    for (int r = 0; r < 8; ++r) s += Bg[i*8+r] * sRi[r*8+l];
    ws[WS_BRI + idx] = s;
  }
  __syncthreads();
  for (int idx = tid; idx < 4096; idx += 256) {
    const int i = idx >> 6, j = idx & 63;
    float s = 0.0f;
    for (int r = 0; r < 8; ++r) s += ws[WS_BRI + i*8+r] * Bg[j*8+r];
    ws[WS_BRIBT + idx] = s;
  }
}

// ---------------- kernel 2: backward LQR Riccati (WMMA, LDS) -------
static __device__ inline void eval_ric(float* ka, const float* X, const float* sA,
                                       const float* sBB, const float* Qg, float* Wt)
{
  mm64(Wt, X, sBB, false, false, 1.0f, 0.0f);   // Wt = y B Ri B^T
  mm64(ka, Wt, X, false, false, 1.0f, 0.0f);    // ka = y B Ri B^T y
  mm64(ka, X, sA, false, false, -1.0f, 1.0f);   // ka -= y A
  mm64(ka, sA, X, true,  false, -1.0f, 1.0f);   // ka -= A^T y
  nn1(Qg);
  __syncthreads();
  for (int i = threadIdx.x; i < 4096; i += 256) ka[i] -= Qg[i];
  __syncthreads();
}
__global__ void __launch_bounds__(256) k_riccati(const float* Fg, const float* Qg, const float* Ag, float* ws)
{
  __shared__ __align__(16) float sA[4096], sBB[4096], y0[4096], yt[4096], ka[4096], Wt[4096];
  float* Ps  = ws + WS_PS;
  float* acc = ws + TM(0);
  asyncCopy16k(sA, Ag);
  asyncCopy16k(sBB, ws + WS_BRIBT);
  for (int i = threadIdx.x; i < 4096; i += 256) { const float v = Fg[i]; y0[i] = v; Ps[(size_t)NTC*4096 + i] = v; }
  asyncWait();
  __syncthreads();
  const float dt = -DT0C;
  for (int s = 0; s < NTC; ++s) {
    eval_ric(ka, y0, sA, sBB, Qg, Wt);
    ew_accum(acc, ka, 1.0f, true);
    ew_stage(yt, y0, ka, 0.5f*dt);
    eval_ric(ka, yt, sA, sBB, Qg, Wt);
    ew_accum(acc, ka, 2.0f, false);
    ew_stage(yt, y0, ka, 0.5f*dt);
    eval_ric(ka, yt, sA, sBB, Qg, Wt);
    ew_accum(acc, ka, 2.0f, false);
    ew_stage(yt, y0, ka, dt);
    eval_ric(ka, yt, sA, sBB, Qg, Wt);
    ew_accum(acc, ka, 1.0f, false);
    ew_stage(y0, y0, acc, dt/6.0f);
    for (int i = threadIdx.x; i < 4096; i += 256) Ps[(size_t)(NTC-1-s)*4096 + i] = y0[i];
    __syncthreads();
  }
}

// ---------------- kernel 3: trajectories for iterations 0 and 1 ----
static __device__ inline void eval_init(float* kv, const float* Xv, const float* Ag,
                                        const float* BB, const float* Pk, float* tv)
{
  mv64(tv, Pk, Xv, false, 1.0f, 0.0f);
  mv64(kv, Ag, Xv, false, 1.0f, 0.0f);
  mv64(kv, BB, tv, false, -1.0f, 1.0f);
}
__global__ void __launch_bounds__(256) k_traj01(const float* Ag, const float* x0g, float* ws, float* outp)
{
  __shared__ __align__(16) float Pk[4096], sTmp[4096];
  __shared__ float y0v[64], ytv[64], kav[64], accv[64], tv[64];
  const float* BB = ws + WS_BRIBT;
  const float* Ps = ws + WS_PS;
  const int tid = threadIdx.x;
  // iteration 0: uncontrolled dynamics
  if (tid < 64) y0v[tid] = x0g[tid];
  __syncthreads();
  for (int i = 0; i < NTC; ++i) {
    if (tid < 64) outp[OUT_X + (0*NTC + i)*64 + tid] = y0v[tid];
    __syncthreads();
    if (i == NTC-1) break;
    mv64(kav, Ag, y0v, false, 1.0f, 0.0f);
    v_accum(accv, kav, 1.0f, true);
    v_stage(ytv, y0v, kav, 0.5f*DT0C);
    mv64(kav, Ag, ytv, false, 1.0f, 0.0f);
    v_accum(accv, kav, 2.0f, false);
    v_stage(ytv, y0v, kav, 0.5f*DT0C);
    mv64(kav, Ag, ytv, false, 1.0f, 0.0f);
    v_accum(accv, kav, 2.0f, false);
    v_stage(ytv, y0v, kav, DT0C);
    mv64(kav, Ag, ytv, false, 1.0f, 0.0f);
    v_accum(accv, kav, 1.0f, false);
    v_stage(y0v, y0v, accv, DT0C/6.0f);
  }
  __syncthreads();
  // iteration 1: LQR initial guess
  if (tid < 64) y0v[tid] = x0g[tid];
  __syncthreads();
  for (int i = 0; i < NTC; ++i) {
    if (tid < 64) { const float v = y0v[tid]; outp[OUT_X + (1*NTC + i)*64 + tid] = v; ws[WS_XG + i*64 + tid] = v; }
    __syncthreads();
    if (i == NTC-1) break;
    loadP_lds(Pk, sTmp, Ps, 2*i);
    eval_init(kav, y0v, Ag, BB, Pk, tv);
    v_accum(accv, kav, 1.0f, true);
    v_stage(ytv, y0v, kav, 0.5f*DT0C);
    loadP_lds(Pk, sTmp, Ps, 2*i+1);
    eval_init(kav, ytv, Ag, BB, Pk, tv);
    v_accum(accv, kav, 2.0f, false);
    v_stage(ytv, y0v, kav, 0.5f*DT0C);
    eval_init(kav, ytv, Ag, BB, Pk, tv);
    v_accum(accv, kav, 2.0f, false);
    v_stage(ytv, y0v, kav, DT0C);
    loadP_lds(Pk, sTmp, Ps, 2*i+2);
    eval_init(kav, ytv, Ag, BB, Pk, tv);
    v_accum(accv, kav, 1.0f, false);
    v_stage(y0v, y0v, accv, DT0C/6.0f);
  }
}

// ---------------- kernel 4: u1 (parallel over time) ----------------
__global__ void __launch_bounds__(64) k_u1(const float* Bg, float* ws, float* outp)
{
  __shared__ float xv[64], vv[64], gv[8], sRi[64];
  const int i = blockIdx.x;
  const int tid = threadIdx.x;
  const float* Ps = ws + WS_PS;
  xv[tid]  = ws[WS_XG + i*64 + tid];
  sRi[tid] = ws[WS_RI + tid];
  __syncthreads();
  {
    const float* P = Ps + (size_t)i*4096;
    float s = 0.0f;
    for (int j = 0; j < 64; ++j) s += P[tid*64 + j] * xv[j];
    vv[tid] = s;
  }
  __syncthreads();
  if (tid < 8) {
    float c = 0.0f;
    for (int n = 0; n < 64; ++n) c += Bg[n*8 + tid] * vv[n];
    gv[tid] = c;
  }
  __syncthreads();
  if (tid < 8) {
    float s = 0.0f;
    #pragma unroll
    for (int l = 0; l < 8; ++l) s += sRi[tid*8 + l] * gv[l];
    outp[OUT_U + (1*NTC + i)*8 + tid] = -s;
    outp[OUT_U + (0*NTC + i)*8 + tid] = 0.0f;  // u_res[0] = 0
  }
}

// ---------------- kernel 5: x_k1 forward (A_k field, WMMA, LDS) ----
__global__ void __launch_bounds__(256) k_xk1(const float* Ag, const float* Bg, const float* Hg,
                                             const float* x0g, float* ws)
{
  __shared__ __align__(16) float Pk[4096], S[4096], Akt[4096], sTmp[4096];
  __shared__ float sB[512], sBRi[512], sRi[64];
  __shared__ float xk[64], pk[64], cd[16], Hx[512], Bt[512], Ep[512], Us[512];
  __shared__ float y0v[64], ytv[64], kav[64], accv[64];
  const float* Ps = ws + WS_PS;
  const float* xg = ws + WS_XG;
  const int tid = threadIdx.x;
  for (int i = tid; i < 512; i += 256) { sB[i] = Bg[i]; sBRi[i] = ws[WS_BRI + i]; }
  if (tid < 64) { sRi[tid] = ws[WS_RI + tid]; y0v[tid] = x0g[tid]; }
  __syncthreads();
  for (int i = 0; i < NTC; ++i) {
    if (tid < 64) ws[WS_XK1 + i*64 + tid] = y0v[tid];
    __syncthreads();
    if (i == NTC-1) break;
    // A_k at t (h = 2i)
    loadX(xk, xg, 2*i); loadP_lds(Pk, sTmp, Ps, 2*i);
    mv64(pk, Pk, xk, false, 1.0f, 0.0f);
    build_Akt_tilde(Akt, Hg, Ag, sB, sBRi, sRi, pk, Ep, cd);
    build_S(S, Hg, sB, sBRi, sRi, xk, Hx, Bt, Us);
    mm64(Akt, S, Pk, false, false, -1.0f, 1.0f);
    mv64(kav, Akt, y0v, false, 1.0f, 0.0f);
    v_accum(accv, kav, 1.0f, true);
    v_stage(ytv, y0v, kav, 0.5f*DT0C);
    // A_k at t+dt/2 (h = 2i+1), shared by k2 and k3
    loadX(xk, xg, 2*i+1); loadP_lds(Pk, sTmp, Ps, 2*i+1);
    mv64(pk, Pk, xk, false, 1.0f, 0.0f);
    build_Akt_tilde(Akt, Hg, Ag, sB, sBRi, sRi, pk, Ep, cd);
    build_S(S, Hg, sB, sBRi, sRi, xk, Hx, Bt, Us);
    mm64(Akt, S, Pk, false, false, -1.0f, 1.0f);
    mv64(kav, Akt, ytv, false, 1.0f, 0.0f);
    v_accum(accv, kav, 2.0f, false);
    v_stage(ytv, y0v, kav, 0.5f*DT0C);
    mv64(kav, Akt, ytv, false, 1.0f, 0.0f);
    v_accum(accv, kav, 2.0f, false);
    v_stage(ytv, y0v, kav, DT0C);
    // A_k at t+dt (h = 2i+2)
    loadX(xk, xg, 2*i+2); loadP_lds(Pk, sTmp, Ps, 2*i+2);
    mv64(pk, Pk, xk, false, 1.0f, 0.0f);
    build_Akt_tilde(Akt, Hg, Ag, sB, sBRi, sRi, pk, Ep, cd);
    build_S(S, Hg, sB, sBRi, sRi, xk, Hx, Bt, Us);
    mm64(Akt, S, Pk, false, false, -1.0f, 1.0f);
    mv64(kav, Akt, ytv, false, 1.0f, 0.0f);
    v_accum(accv, kav, 1.0f, false);
    v_stage(y0v, y0v, accv, DT0C/6.0f);
  }
}

// ---------------- kernel 6: backward P_k1 (WMMA, LDS) --------------
static __device__ inline void build_AktQk(int h,
    float* Akt, float* S, float* Pk, float* M1, float* Qk, float* ka, float* sTmp,
    const float* Ps, const float* xg, const float* Hg, const float* Ag,
    const float* sB, const float* sBRi, const float* sRi,
    float* xk, float* pk, float* cd, float* Hx, float* Bt, float* Ep, float* Us)
{
  loadX(xk, xg, h);
  loadP_lds(Pk, sTmp, Ps, h);
  mv64(pk, Pk, xk, false, 1.0f, 0.0f);
  build_Akt_tilde(Akt, Hg, Ag, sB, sBRi, sRi, pk, Ep, cd);
  build_S(S, Hg, sB, sBRi, sRi, xk, Hx, Bt, Us);
  // M1 = (Ep Ri) Ep^T ;  Qk = M1 + M1^T - Pk S Pk
  mk8_XR(Us, Ep, sRi);
  mk8_XYt(M1, Us, Ep, 1.0f, 0.0f);
  mm64(ka, Pk, S, false, false, 1.0f, 0.0f);
  mm64(Qk, ka, Pk, false, false, -1.0f, 0.0f);
  __syncthreads();
  for (int idx = threadIdx.x; idx < 4096; idx += 256) {
    const int i = idx >> 6, j = idx & 63;
    Qk[idx] += M1[idx] + M1[j*64 + i];
  }
  __syncthreads();
  mm64(Akt, S, Pk, false, false, -1.0f, 1.0f);  // A_k = A_k_tilde - S Pk
}
static __device__ inline void eval_Pfield(float* ka, const float* X, const float* Akt, const float* Qk)
{
  mm64(ka, X, Akt, false, false, -1.0f, 0.0f);
  mm64(ka, Akt, X, true,  false, -1.0f, 1.0f);
  nn1(Qk);
  __syncthreads();
  for (int i = threadIdx.x; i < 4096; i += 256) ka[i] -= Qk[i];
  __syncthreads();
}
__global__ void __launch_bounds__(256) k_Pk1(const float* Fg, const float* Ag, const float* Bg,
                                             const float* Hg, float* ws)
{
  __shared__ __align__(16) float Pk[4096], S[4096], Akt[4096], y0[4096], yt[4096], ka[4096], sTmp[4096];
  __shared__ float sB[512], sBRi[512], sRi[64];
  __shared__ float xk[64], pk[64], cd[16], Hx[512], Bt[512], Ep[512], Us[512];
  float* M1  = ws + TM(0);
  float* Qk  = ws + TM(1);
  float* acc = ws + TM(2);
  float* Pdst = ws + WS_PK1;
  const float* Ps = ws + WS_PS;
  const float* xg = ws + WS_XG;
  const int tid = threadIdx.x;
  for (int i = tid; i < 512; i += 256) { sB[i] = Bg[i]; sBRi[i] = ws[WS_BRI + i]; }
  if (tid < 64) sRi[tid] = ws[WS_RI + tid];
  for (int i = tid; i < 4096; i += 256) { const float v = Fg[i]; y0[i] = v; Pdst[(size_t)NTC*4096 + i] = v; }
  __syncthreads();
  const float dt = -DT0C;
  for (int s = 0; s < NTC; ++s) {
    const int h0 = 2*(NTC - s);
    build_AktQk(h0,   Akt, S, Pk, M1, Qk, ka, sTmp, Ps, xg, Hg, Ag, sB, sBRi, sRi, xk, pk, cd, Hx, Bt, Ep, Us);
    eval_Pfield(ka, y0, Akt, Qk);
    ew_accum(acc, ka, 1.0f, true);
    ew_stage(yt, y0, ka, 0.5f*dt);
    build_AktQk(h0-1, Akt, S, Pk, M1, Qk, ka, sTmp, Ps, xg, Hg, Ag, sB, sBRi, sRi, xk, pk, cd, Hx, Bt, Ep, Us);
    eval_Pfield(ka, yt, Akt, Qk);
    ew_accum(acc, ka, 2.0f, false);
    ew_stage(yt, y0, ka, 0.5f*dt);
    eval_Pfield(ka, yt, Akt, Qk);          // k3 reuses same A_k, Q_k
    ew_accum(acc, ka, 2.0f, false);
    ew_stage(yt, y0, ka, dt);
    build_AktQk(h0-2, Akt, S, Pk, M1, Qk, ka, sTmp, Ps, xg, Hg, Ag, sB, sBRi, sRi, xk, pk, cd, Hx, Bt, Ep, Us);
    eval_Pfield(ka, yt, Akt, Qk);
    ew_accum(acc, ka, 1.0f, false);
    ew_stage(y0, y0, acc, dt/6.0f);
    for (int i = threadIdx.x; i < 4096; i += 256) Pdst[(size_t)(NTC-1-s)*4096 + i] = y0[i];
    __syncthreads();
  }
}

// ---------------- kernel 7: u_grid (parallel over time) ------------
__global__ void __launch_bounds__(256) k_ugrid(const float* Hg, float* ws, float* outp)
{
  __shared__ float xkv[64], x1v[64], vv[64], Hx[512], gv[8], sRi[64];
  const int i = blockIdx.x;
  const int tid = threadIdx.x;
  if (tid < 64) { xkv[tid] = ws[WS_XG + i*64 + tid]; x1v[tid] = ws[WS_XK1 + i*64 + tid]; sRi[tid] = ws[WS_RI + tid]; }
  __syncthreads();
  const float* P = ws + WS_PK1 + (size_t)i*4096;
  if (tid < 64) {
    float s = 0.0f;
    for (int j = 0; j < 64; ++j) s += P[tid*64 + j] * x1v[j];
    vv[tid] = s;
  }
  for (int idx = tid; idx < 512; idx += 256) {
    const int j = idx >> 3, l = idx & 7;
    float s = 0.0f;
    for (int n = 0; n < 64; ++n) s += Hg[j*512 + n*8 + l] * xkv[n];
    Hx[idx] = s;
  }
  __syncthreads();
  if (tid < 8) {
    float s = 0.0f;
    for (int j = 0; j < 64; ++j) s += Hx[j*8 + tid] * vv[j];
    gv[tid] = s;
  }
  __syncthreads();
  if (tid < 8) {
    float s = 0.0f;
    #pragma unroll
    for (int l = 0; l < 8; ++l) s += sRi[tid*8 + l] * gv[l];
    const float u = -s;
    outp[OUT_U + (2*NTC + i)*8 + tid] = u;
    ws[WS_UG + i*8 + tid] = u;
  }
}

// ---------------- kernel 8: x_eval forward -------------------------
static __device__ inline void build_evalf(int h, float* Akt, float* Pk, float* buv, float* sTmp,
    const float* Ps, const float* xg, const float* ug, const float* Hg, const float* Ag,
    const float* sB, const float* sBRi, const float* sRi,
    float* xk, float* pk, float* cd, float* Hx, float* Bt, float* Ep, float* uu)
{
  loadX(xk, xg, h);
  loadP_lds(Pk, sTmp, Ps, h);
  mv64(pk, Pk, xk, false, 1.0f, 0.0f);
  build_Akt_tilde(Akt, Hg, Ag, sB, sBRi, sRi, pk, Ep, cd);
  h_mid(Hx, Hg, xk);
  __syncthreads();
  for (int i = threadIdx.x; i < 512; i += 256) Bt[i] = sB[i] + Hx[i];
  __syncthreads();
  loadU(uu, ug, h);
  if (threadIdx.x < 64) {
    float s = 0.0f;
    #pragma unroll
    for (int l = 0; l < 8; ++l) s += Bt[threadIdx.x*8 + l] * uu[l];
    buv[threadIdx.x] = s;
  }
  __syncthreads();
}
static __device__ inline void stage_evalf(float* kv, const float* Xv, const float* Akt, const float* buv)
{
  nn3(kv, Xv, Akt); nn1(buv);
  __syncthreads();
  if (threadIdx.x < 64) kv[threadIdx.x] = buv[threadIdx.x];
  __syncthreads();
  mv64(kv, Akt, Xv, false, 1.0f, 1.0f);
}
__global__ void __launch_bounds__(256) k_xeval(const float* Ag, const float* Bg, const float* Hg,
                                               const float* x0g, float* ws, float* outp)
{
  __shared__ __align__(16) float Pk[4096], Akt[4096], sTmp[4096];
  __shared__ float sB[512], sBRi[512], sRi[64];
  __shared__ float xk[64], pk[64], cd[16], Hx[512], Bt[512], Ep[512], uu[8], buv[64];
  __shared__ float y0v[64], ytv[64], kav[64], accv[64];
  const float* Ps = ws + WS_PS;
  const float* xg = ws + WS_XG;
  const float* ug = ws + WS_UG;
  const int tid = threadIdx.x;
  for (int i = tid; i < 512; i += 256) { sB[i] = Bg[i]; sBRi[i] = ws[WS_BRI + i]; }
  if (tid < 64) { sRi[tid] = ws[WS_RI + tid]; y0v[tid] = x0g[tid]; }
  __syncthreads();
  for (int i = 0; i < NTC; ++i) {
    if (tid < 64) outp[OUT_X + (2*NTC + i)*64 + tid] = y0v[tid];
    __syncthreads();
    if (i == NTC-1) break;
    build_evalf(2*i,   Akt, Pk, buv, sTmp, Ps, xg, ug, Hg, Ag, sB, sBRi, sRi, xk, pk, cd, Hx, Bt, Ep, uu);
    stage_evalf(kav, y0v, Akt, buv);
    v_accum(accv, kav, 1.0f, true);
    v_stage(ytv, y0v, kav, 0.5f*DT0C);
    build_evalf(2*i+1, Akt, Pk, buv, sTmp, Ps, xg, ug, Hg, Ag, sB, sBRi, sRi, xk, pk, cd, Hx, Bt, Ep, uu);
    stage_evalf(kav, ytv, Akt, buv);
    v_accum(accv, kav, 2.0f, false);
    v_stage(ytv, y0v, kav, 0.5f*DT0C);
    stage_evalf(kav, ytv, Akt, buv);
    v_accum(accv, kav, 2.0f, false);
    v_stage(ytv, y0v, kav, DT0C);
    build_evalf(2*i+2, Akt, Pk, buv, sTmp, Ps, xg, ug, Hg, Ag, sB, sBRi, sRi, xk, pk, cd, Hx, Bt, Ep, uu);
    stage_evalf(kav, ytv, Akt, buv);
    v_accum(accv, kav, 1.0f, false);
    v_stage(y0v, y0v, accv, DT0C/6.0f);
  }
}

// ---------------- host entry ---------------------------------------
extern "C" void kernel_launch(void* const* d_in, const int* in_sizes, int n_in,
                              void* d_out, int out_size, void* d_ws, size_t ws_size,
                              hipStream_t stream)
{
  (void)in_sizes; (void)n_in; (void)out_size; (void)ws_size;
  const float* Fg  = (const float*)d_in[0];
  const float* Qg  = (const float*)d_in[1];
  const float* Rg  = (const float*)d_in[2];
  const float* Ag  = (const float*)d_in[3];
  const float* Bg  = (const float*)d_in[4];
  const float* Hg  = (const float*)d_in[5];
  const float* x0g = (const float*)d_in[6];
  float* outp = (float*)d_out;
  float* ws   = (float*)d_ws;

  k_prep   <<<1,   256, 0, stream>>>(Rg, Bg, ws);
  k_riccati<<<1,   256, 0, stream>>>(Fg, Qg, Ag, ws);
  k_traj01 <<<1,   256, 0, stream>>>(Ag, x0g, ws, outp);
  k_u1     <<<NTC, 64,  0, stream>>>(Bg, ws, outp);
  k_xk1    <<<1,   256, 0, stream>>>(Ag, Bg, Hg, x0g, ws);
  k_Pk1    <<<1,   256, 0, stream>>>(Fg, Ag, Bg, Hg, ws);
  k_ugrid  <<<NTC, 256, 0, stream>>>(Hg, ws, outp);
  k_xeval  <<<1,   256, 0, stream>>>(Ag, Bg, Hg, x0g, ws, outp);
}